// PairwiseSubtractionLayer_23665269801128
// MI455X (gfx1250) — compile-verified
//
#include <hip/hip_runtime.h>
#include <hip/hip_bf16.h>

// Problem constants (fixed by the reference: A[64,512,2], B[2,4096]).
#define NQ    (64 * 512)   // 32768 queries
#define NPTS  4096         // forbidden points
#define BLOCK 256          // 8 wave32s per block

__global__ __launch_bounds__(BLOCK)
void linf_nn_kernel(const float* __restrict__ A,
                    const float* __restrict__ B,
                    float* __restrict__ out) {
    // LDS copy of B, identity layout: sB[0..NPTS-1]=Bx, sB[NPTS..2*NPTS-1]=By. 32 KB.
    __shared__ float sB[2 * NPTS];

    const int tid = threadIdx.x;

    // ---- Stage B into LDS via CDNA5 async global->LDS (ASYNCcnt path) ----
    {
        // Low 32 bits of a generic LDS pointer are the LDS byte offset.
        unsigned lds_base = (unsigned)(size_t)(&sB[0]);
        // 32768 bytes total, 256 lanes x 16B = 4096 bytes per pass -> 8 passes.
        #pragma unroll
        for (int p = 0; p < (2 * NPTS * 4) / (BLOCK * 16); ++p) {
            unsigned off      = (unsigned)((p * BLOCK + tid) * 16);
            unsigned lds_addr = lds_base + off;
            // GVS mode: vdst = LDS addr (VGPR), vaddr = 32-bit offset (VGPR),
            //           saddr = 64-bit base (SGPR pair)
            asm volatile("global_load_async_to_lds_b128 %0, %1, %2 offset:0"
                         :
                         : "v"(lds_addr), "v"(off), "s"(B)
                         : "memory");
        }
#if __has_builtin(__builtin_amdgcn_s_wait_asynccnt)
        __builtin_amdgcn_s_wait_asynccnt(0);
#else
        asm volatile("s_wait_asynccnt 0x0" ::: "memory");
#endif
    }
    __syncthreads();

    // ---- Per-thread query scan over LDS-resident candidates ----
    const int q = blockIdx.x * BLOCK + tid;       // 0..NQ-1
    const float ax = A[q * 2 + 0];
    const float ay = A[q * 2 + 1];

    const float* __restrict__ sBx = &sB[0];
    const float* __restrict__ sBy = &sB[NPTS];

    float best  = 3.402823466e+38f;
    int   bestn = 0;

    // All lanes read the same candidate index -> LDS broadcast, no conflicts.
    for (int n = 0; n < NPTS; n += 4) {
        const float4 bx = *(const float4*)(sBx + n);
        const float4 by = *(const float4*)(sBy + n);

        const float d0 = fmaxf(fabsf(ax - bx.x), fabsf(ay - by.x));
        const float d1 = fmaxf(fabsf(ax - bx.y), fabsf(ay - by.y));
        const float d2 = fmaxf(fabsf(ax - bx.z), fabsf(ay - by.z));
        const float d3 = fmaxf(fabsf(ax - bx.w), fabsf(ay - by.w));

        // Strict '<' in ascending n keeps the FIRST minimum (matches jnp.argmin).
        if (d0 < best) { best = d0; bestn = n + 0; }
        if (d1 < best) { best = d1; bestn = n + 1; }
        if (d2 < best) { best = d2; bestn = n + 2; }
        if (d3 < best) { best = d3; bestn = n + 3; }
    }

    out[q * 2 + 0] = ax - sBx[bestn];
    out[q * 2 + 1] = ay - sBy[bestn];
}

extern "C" void kernel_launch(void* const* d_in, const int* in_sizes, int n_in,
                              void* d_out, int out_size, void* d_ws, size_t ws_size,
                              hipStream_t stream) {
    const float* A = (const float*)d_in[0];   // [64, 512, 2] fp32
    const float* B = (const float*)d_in[1];   // [2, 4096]   fp32
    float* out     = (float*)d_out;           // [64, 512, 2] fp32

    (void)in_sizes; (void)n_in; (void)out_size; (void)d_ws; (void)ws_size;

    dim3 grid(NQ / BLOCK);   // 128 blocks
    dim3 block(BLOCK);       // 256 threads = 8 wave32s
    linf_nn_kernel<<<grid, block, 0, stream>>>(A, B, out);
}